// PWLNNFcn_41059887349838
// MI455X (gfx1250) — compile-verified
//
#include <hip/hip_runtime.h>

// Problem constants (from reference setup_inputs)
#define NSMP 4096
#define NCTR 2048
#define DIN  64
#define DOUT 64
#define KNN  4

typedef float v2f __attribute__((ext_vector_type(2)));
typedef float v8f __attribute__((ext_vector_type(8)));

#define FLT_BIG 3.402823466e38f

// ---------------------------------------------------------------------------
// Kernel 0: center squared norms cc[j] = sum_i c[j,i]^2
// ---------------------------------------------------------------------------
__global__ void cc_kernel(const float* __restrict__ ctrs, float* __restrict__ cc) {
    int j = blockIdx.x * blockDim.x + threadIdx.x;
    if (j < NCTR) {
        const float* c = ctrs + (size_t)j * DIN;
        float s = 0.f;
#pragma unroll 8
        for (int i = 0; i < DIN; ++i) s += c[i] * c[i];
        cc[j] = s;
    }
}

// ---------------------------------------------------------------------------
// Kernel 1: per-center bias b[j,o] = offsets[j,o] - sum_i W[j,i,o]*c[j,i]
// (hoists the 'knio,kni->no' einsum out of the per-sample path)
// ---------------------------------------------------------------------------
__global__ void bias_kernel(const float* __restrict__ wts,
                            const float* __restrict__ ctrs,
                            const float* __restrict__ offsets,
                            float* __restrict__ b) {
    int t = blockIdx.x * blockDim.x + threadIdx.x;   // 2048*64 threads
    int j = t >> 6;
    int o = t & 63;
    const float* W = wts + (size_t)j * DIN * DOUT + o;  // stride DOUT over i
    const float* c = ctrs + (size_t)j * DIN;
    float s = 0.f;
#pragma unroll 8
    for (int i = 0; i < DIN; ++i) s += W[(size_t)i * DOUT] * c[i];
    b[t] = offsets[t] - s;
}

// ---------------------------------------------------------------------------
// Kernel 2: fp32 WMMA GEMM  S[n,j] = cc[j] - 2 * (x_n . c_j)
// (|x_n|^2 is row-constant -> irrelevant for per-row top-k, so omitted)
// One wave per 16x16 tile; 16 x V_WMMA_F32_16X16X4_F32 over K=64.
// 8 waves/block share the same 16-row X slab across 8 adjacent col tiles.
// ---------------------------------------------------------------------------
__global__ void score_wmma_kernel(const float* __restrict__ x,
                                  const float* __restrict__ ctrs,
                                  const float* __restrict__ cc,
                                  float* __restrict__ S) {
    const int wave = threadIdx.x >> 5;
    const int lane = threadIdx.x & 31;
    const int rt = blockIdx.x >> 4;          // 0..255  row tile (16 samples)
    const int cg = blockIdx.x & 15;          // col group of 8 tiles
    const int ct = cg * 8 + wave;            // 0..127  col tile (16 centers)
    const int rowBase = rt * 16;
    const int colBase = ct * 16;

    // A 16x4 f32 layout: lane&15 = M row, (lane>>4)*2 = K pair base
    const int r    = lane & 15;
    const int koff = (lane >> 4) * 2;
    const float* ax = x    + (size_t)(rowBase + r) * DIN + koff;
    const float* bx = ctrs + (size_t)(colBase + r) * DIN + koff;  // B = C^T tile

    v8f acc = {};
#pragma unroll
    for (int kk = 0; kk < DIN; kk += 4) {
        v2f a; a.x = ax[kk]; a.y = ax[kk + 1];
        v2f bb; bb.x = bx[kk]; bb.y = bx[kk + 1];
        acc = __builtin_amdgcn_wmma_f32_16x16x4_f32(
            /*neg_a=*/false, a, /*neg_b=*/false, bb,
            /*c_mod=*/(short)0, acc, /*reuse_a=*/false, /*reuse_b=*/false);
    }

    // C/D layout: lane holds column N = lane&15; rows M = v + 8*(lane>>4)
    const int col   = colBase + (lane & 15);
    const float ccv = cc[col];
    const int rbase = rowBase + ((lane >> 4) << 3);
#pragma unroll
    for (int v = 0; v < 8; ++v) {
        S[(size_t)(rbase + v) * NCTR + col] = ccv - 2.0f * acc[v];
    }
}

// ---------------------------------------------------------------------------
// Kernel 3: per-row top-4 (smallest). One wave32 per row; lane-strided
// coalesced scan -> sorted local top-4 -> 4-round cross-lane argmin merge.
// ---------------------------------------------------------------------------
__global__ void topk_kernel(const float* __restrict__ S, int* __restrict__ topidx) {
    const int wave = threadIdx.x >> 5;
    const int lane = threadIdx.x & 31;
    const int n = blockIdx.x * 8 + wave;
    const float* row = S + (size_t)n * NCTR;

    float bv0 = FLT_BIG, bv1 = FLT_BIG, bv2 = FLT_BIG, bv3 = FLT_BIG;
    int   bi0 = -1, bi1 = -1, bi2 = -1, bi3 = -1;

    for (int j = lane; j < NCTR; j += 32) {
        float v = row[j];
        if (v < bv3) {
            if (v < bv2) {
                bv3 = bv2; bi3 = bi2;
                if (v < bv1) {
                    bv2 = bv1; bi2 = bi1;
                    if (v < bv0) { bv1 = bv0; bi1 = bi0; bv0 = v; bi0 = j; }
                    else         { bv1 = v;  bi1 = j; }
                } else { bv2 = v; bi2 = j; }
            } else { bv3 = v; bi3 = j; }
        }
    }

    // Merge: extract global min 4 times; winning lane pops its sorted list.
    for (int t = 0; t < KNN; ++t) {
        float cand = bv0;
        int   cidx = bi0;
        int   clane = lane;
        for (int off = 16; off > 0; off >>= 1) {
            float ov = __shfl_down(cand, off, 32);
            int   oi = __shfl_down(cidx, off, 32);
            int   ol = __shfl_down(clane, off, 32);
            if (ov < cand) { cand = ov; cidx = oi; clane = ol; }
        }
        int wi = __shfl(cidx, 0, 32);
        int wl = __shfl(clane, 0, 32);
        if (lane == wl) {  // pop front of sorted list
            bv0 = bv1; bi0 = bi1;
            bv1 = bv2; bi1 = bi2;
            bv2 = bv3; bi2 = bi3;
            bv3 = FLT_BIG; bi3 = -1;
        }
        if (lane == 0) topidx[n * KNN + t] = wi;
    }
}

// ---------------------------------------------------------------------------
// Kernel 4: out[n,o] = sum_k [ (W_{j_k}^T x_n)[o] + b[j_k,o] ]
// 256 threads = 4 samples x 64 outputs; x_n staged in LDS; W reads are
// coalesced over o and L2-resident (wts = 32 MB < 192 MB L2).
// ---------------------------------------------------------------------------
__global__ void output_kernel(const float* __restrict__ x,
                              const float* __restrict__ wts,
                              const float* __restrict__ b,
                              const int* __restrict__ topidx,
                              float* __restrict__ out) {
    __shared__ float xs[4][DIN];
    __shared__ int   js[4][KNN];
    const int s = threadIdx.x >> 6;   // sample slot in block
    const int o = threadIdx.x & 63;
    const int n = blockIdx.x * 4 + s;

    xs[s][o] = x[(size_t)n * DIN + o];
    if (o < KNN) js[s][o] = topidx[n * KNN + o];
    __syncthreads();

    float acc = 0.f;
#pragma unroll
    for (int k = 0; k < KNN; ++k) {
        const int j = js[s][k];
        const float* W = wts + (size_t)j * DIN * DOUT + o;
        acc += b[(size_t)j * DOUT + o];
        float a = 0.f;
#pragma unroll 8
        for (int i = 0; i < DIN; ++i) a += W[(size_t)i * DOUT] * xs[s][i];
        acc += a;
    }
    out[(size_t)n * DOUT + o] = acc;
}

// ---------------------------------------------------------------------------
extern "C" void kernel_launch(void* const* d_in, const int* in_sizes, int n_in,
                              void* d_out, int out_size, void* d_ws, size_t ws_size,
                              hipStream_t stream) {
    const float* x       = (const float*)d_in[0];
    const float* ctrs    = (const float*)d_in[1];
    const float* wts     = (const float*)d_in[2];
    const float* offsets = (const float*)d_in[3];
    float* out = (float*)d_out;

    // Workspace layout (~33 MB): S | cc | b | topidx
    float* S      = (float*)d_ws;                    // NSMP*NCTR
    float* cc     = S + (size_t)NSMP * NCTR;         // NCTR
    float* b      = cc + NCTR;                       // NCTR*DOUT
    int*   topidx = (int*)(b + (size_t)NCTR * DOUT); // NSMP*KNN

    cc_kernel   <<<NCTR / 256, 256, 0, stream>>>(ctrs, cc);
    bias_kernel <<<(NCTR * DOUT) / 256, 256, 0, stream>>>(wts, ctrs, offsets, b);
    score_wmma_kernel<<<(NSMP / 16) * (NCTR / 16) / 8, 256, 0, stream>>>(x, ctrs, cc, S);
    topk_kernel <<<NSMP / 8, 256, 0, stream>>>(S, topidx);
    output_kernel<<<NSMP / 4, 256, 0, stream>>>(x, wts, b, topidx, out);
}